// torch_forward_77154792505836
// MI455X (gfx1250) — compile-verified
//
#include <hip/hip_runtime.h>
#include <stdint.h>

// ---- reference constants ----
#define DTW       0.5f
#define VRDT      0.05f        // VR*DT
#define RECI      2            // record every 2 steps
#define NRECS     250
#define TOUT      251          // output columns per batch element
#define BATCHN    256
#define LANES     32           // wave32-native block
#define NBLK      (BATCHN/LANES)
#define CHUNK_REC 50           // records buffered in LDS per flush
#define NCHUNK    (NRECS/CHUNK_REC)

// Low 32 bits of a generic pointer into LDS == workgroup-relative LDS byte address
__device__ __forceinline__ uint32_t lds_off(const void* p) {
  return (uint32_t)(uintptr_t)p;
}
// CDNA5 async DMA: global -> LDS (tracked by ASYNCcnt)
__device__ __forceinline__ void async_g2l_b32(uint32_t loff, const float* g) {
  asm volatile("global_load_async_to_lds_b32 %0, %1, off" :: "v"(loff), "v"(g) : "memory");
}
// CDNA5 async DMA: LDS -> global (tracked by ASYNCcnt)
__device__ __forceinline__ void async_l2g_b32(float* g, uint32_t loff) {
  asm volatile("global_store_async_from_lds_b32 %0, %1, off" :: "v"(g), "v"(loff) : "memory");
}
__device__ __forceinline__ void wait_async0() {
#if __has_builtin(__builtin_amdgcn_s_wait_asynccnt)
  __builtin_amdgcn_s_wait_asynccnt(0);
#else
  asm volatile("s_wait_asynccnt 0x0" ::: "memory");
#endif
}
__device__ __forceinline__ void wait_ds0() {
#if __has_builtin(__builtin_amdgcn_s_wait_dscnt)
  __builtin_amdgcn_s_wait_dscnt(0);
#else
  asm volatile("s_wait_dscnt 0x0" ::: "memory");
#endif
}
// Single v_rcp_f32 (TRANS op, co-executes with VALU): 1-ulp reciprocal,
// avoids the ~10-op exact-division expansion on the serial critical path.
__device__ __forceinline__ float fast_rcp(float x) {
  return __builtin_amdgcn_rcpf(x);
}

__global__ __launch_bounds__(LANES) void film_growth_kernel(
    const float* __restrict__ z, float* __restrict__ out)
{
  __shared__ float zs[3*LANES];
  __shared__ float hist[CHUNK_REC][LANES];

  const int tid = threadIdx.x;
  const int b0  = blockIdx.x * LANES;   // first batch element of this block
  const int gb  = b0 + tid;             // this lane's batch element

  // ---- stage this block's z slice (96 contiguous floats) via async DMA ----
  {
    const float* gsrc = z + b0*3;
    const uint32_t lbase = lds_off(&zs[0]);
#pragma unroll
    for (int q = 0; q < 3; ++q)
      async_g2l_b32(lbase + (uint32_t)((q*LANES + tid)*4), gsrc + q*LANES + tid);
    wait_async0();
    __syncthreads();
  }

  const float Cv   = exp10f(-zs[3*tid+0]);              // 10^{-z0}
  const float Kp43 = powf(zs[3*tid+1], 4.0f/3.0f);      // K^{4/3}
  const float jmin = zs[3*tid+2];

  // Qmin is set at i==1 with R_prev = R_FILM0 = 1:
  //   j1 = BC1/(R0+1) = 0.05/2,  beta1 = j1/DT
  const float j1    = VRDT / 2.0f;
  const float beta1 = j1 / DTW;
  const float Qmin  = cbrtf(81.0f / (128.0f * beta1)) * Kp43;

  float Q = 0.0f, h = 0.0f, R = 1.0f;

  out[(size_t)gb*TOUT] = 0.0f;          // t = 0 column is zero

  int i = 1;
  for (int c = 0; c < NCHUNK; ++c) {
    for (int r = 0; r < CHUNK_REC; ++r) {
#pragma unroll
      for (int s = 0; s < RECI; ++s, ++i) {
        const float BC = VRDT * (float)i;
        // Closed form of the 254-step Thomas scan (telescoping Mobius product,
        // double characteristic root):  j = J_FAC*BC*(1+cp0)/(1+254(1+cp0))
        //                                 = BC/(R + 255*h_x) = BC/(R+1)
        const float j  = BC * fast_rcp(R + 1.0f);
        Q = fmaf(j, DTW, Q);
        const float x = 1.0e8f * (Q - Qmin);
        const float w = fast_rcp(1.0f + __expf(-x));    // saturating sigmoid gate
        const float dh = Cv * (j - jmin) * DTW;
        h = w * fmaxf(h + dh, 0.0f) + (1.0f - w) * h;
        const float rho = w * fmaxf(8.0e6f * __expf(-0.1f * j), 2.0e6f);
        R = w * fmaxf(fmaf(rho * (j - jmin) * Cv, DTW, R), 0.0f) + (1.0f - w) * R;
      }
      hist[r][tid] = h;                  // record h at even steps
    }
    // ---- flush chunk: LDS -> global via async DMA, coalesced over t ----
    wait_ds0();                          // LDS writes visible before DMA reads
    __syncthreads();
    for (int e = tid; e < LANES*CHUNK_REC; e += LANES) {
      const int bl = e / CHUNK_REC;      // local batch element
      const int r  = e - bl*CHUNK_REC;   // record within chunk
      float* g = out + (size_t)(b0 + bl)*TOUT + (c*CHUNK_REC + r + 1);
      async_l2g_b32(g, lds_off(&hist[r][bl]));
    }
    wait_async0();                       // LDS reusable next chunk
    __syncthreads();
  }
}

extern "C" void kernel_launch(void* const* d_in, const int* /*in_sizes*/, int /*n_in*/,
                              void* d_out, int /*out_size*/, void* /*d_ws*/, size_t /*ws_size*/,
                              hipStream_t stream)
{
  const float* z = (const float*)d_in[0];   // (256, 3) float32
  float* out = (float*)d_out;               // (256, 251) float32
  film_growth_kernel<<<dim3(NBLK), dim3(LANES), 0, stream>>>(z, out);
}